// KNNMutationSite_83708912599258
// MI455X (gfx1250) — compile-verified
//
#include <hip/hip_runtime.h>
#include <stdint.h>

// ---------------------------------------------------------------------------
// KNN mutation-site mask, fused single kernel for MI455X (gfx1250, wave32).
//
// G=128 graphs x NPG=2048 nodes, <=2 centers/graph, K=64 smallest per-graph
// distances -> boolean mask. Total traffic ~8MB => <1us at 23.3 TB/s; the win
// is a single launch, one WGP per graph, all intermediates LDS-resident
// (~10KB of the 320KB/WGP pool).
//
// CDNA5 paths:
//   * V_WMMA_F32_16X16X4_F32 : node(16) x center(16) dot tiles, K=3 coords
//     zero-padded to 4. A (16x4 f32): VGPR0={K0|K2}, VGPR1={K1|K3} across
//     lane halves; B mirrored. C/D: VGPR j holds rows j (lanes 0-15) and
//     j+8 (lanes 16-31), one center column per lane.
//   * ds_swizzle_b32 (immediate XOR patterns) for branchless wave32 min
//     reductions, batched 8-wide so the compiler can overlap DS latency via
//     partial s_wait_dscnt counts.
//   * ballot + ffs (per 16-lane half) for the lowest-index argmin: replaces a
//     second 4-step swizzle reduction with one v_cmp + SALU bit ops.
//   * global_prefetch_b8 via __builtin_prefetch for the next tile.
//
// Exactness: distance to the argmin center is RECOMPUTED as sum((x-c)^2)
// (same as the reference); top-K is an exact radix-select on the raw float
// bits (monotonic for non-negative floats) with lowest-index tie-break,
// matching jax.lax.top_k semantics.
// ---------------------------------------------------------------------------

typedef __attribute__((ext_vector_type(2))) float v2f;
typedef __attribute__((ext_vector_type(8))) float v8f;

#define NPG    2048      // nodes per graph (reference constant)
#define BLOCK  256       // 8 wave32 waves
#define KSEL   64        // top-K
#define MAXC   16        // max centers/graph supported (ref uses 2)
#define BIGF   1e30f

// Branchless lane exchange: ds_swizzle_b32 group-of-32, lane' = lane ^ XORMASK.
template <int XORMASK>
__device__ __forceinline__ float swz_f(float v) {
  return __int_as_float(
      __builtin_amdgcn_ds_swizzle(__float_as_int(v), (XORMASK << 10) | 0x1f));
}

__global__ __launch_bounds__(BLOCK)
void knn_mutation_kernel(const float* __restrict__ pos,      // [N,3] f32
                         const uint8_t* __restrict__ is_mut, // [N] bool (1B)
                         const int* __restrict__ atom_name,  // [N] i32
                         const int* __restrict__ ca_id_p,    // [1] i32
                         float* __restrict__ out)            // [N] mask
{
  __shared__ float    s_dist[NPG];          // 8 KB
  __shared__ float    s_cx[MAXC], s_cy[MAXC], s_cz[MAXC], s_c2[MAXC];
  __shared__ unsigned s_mutbits[NPG / 32];  // 256 B
  __shared__ int      s_ncent;
  __shared__ unsigned s_hist[256];          // 1 KB (radix select)
  __shared__ unsigned s_scan[BLOCK];        // 1 KB (tie-rank scan)
  __shared__ unsigned s_bcast[2];

  const int g     = blockIdx.x;
  const int tid   = threadIdx.x;
  const int gbase = g * NPG;
  const int ca    = ca_id_p[0];

  // ---- Phase A: mutation bitmask + center compaction (order irrelevant:
  //      only the node mask is returned, never center indices) -------------
  if (tid == 0) s_ncent = 0;
  for (int i = tid; i < NPG / 32; i += BLOCK) s_mutbits[i] = 0u;
  if (tid < MAXC) { s_cx[tid] = 0.f; s_cy[tid] = 0.f; s_cz[tid] = 0.f; s_c2[tid] = 0.f; }
  __syncthreads();

  for (int r = tid; r < NPG; r += BLOCK) {
    const float* p = pos + (size_t)(gbase + r) * 3;
    float px = p[0], py = p[1], pz = p[2];
    bool m = (is_mut[gbase + r] != 0) && (atom_name[gbase + r] == ca);
    if (m) {
      atomicOr(&s_mutbits[r >> 5], 1u << (r & 31));
      int slot = atomicAdd(&s_ncent, 1);
      if (slot < MAXC) {
        s_cx[slot] = px; s_cy[slot] = py; s_cz[slot] = pz;
        s_c2[slot] = px * px + py * py + pz * pz;
      }
    }
  }
  __syncthreads();
  int ncent = s_ncent; if (ncent > MAXC) ncent = MAXC;

  // ---- Phase B: WMMA nearest-center distance, 16 rows per tile -----------
  const int lane = tid & 31;
  const int wave = tid >> 5;
  const int n    = lane & 15;   // center column owned by this lane
  const int hi   = lane >> 4;   // half-wave: selects K={0,1} vs K={2,3}

  // B (4x16 f32): VGPR0 = K0|K2 rows, VGPR1 = K1|K3 rows; unused columns are
  // zero-position centers (dot = 0), masked below via c2lane = +BIG.
  v2f bmat;
  bmat.x = hi ? s_cz[n] : s_cx[n];
  bmat.y = hi ? 0.0f    : s_cy[n];
  const bool  valid  = (n < ncent);
  const float c2lane = valid ? s_c2[n] : BIGF;

  for (int t = wave; t < NPG / 16; t += BLOCK / 32) {
    const int base = t * 16;
    const int arow = base + n;  // A-matrix row supplied by this lane
    const float* p = pos + (size_t)(gbase + arow) * 3;
    if (t + (BLOCK / 32) < NPG / 16)  // global_prefetch_b8 next tile
      __builtin_prefetch(pos + (size_t)(gbase + arow + 16 * (BLOCK / 32)) * 3, 0, 1);
    float px = p[0], py = p[1], pz = p[2];
    v2f amat;
    amat.x = hi ? pz   : px;
    amat.y = hi ? 0.0f : py;    // K=3 zero-padded

    v8f acc = {};
    acc = __builtin_amdgcn_wmma_f32_16x16x4_f32(false, amat, false, bmat,
                                                (short)0, acc, false, false);

    // Per-row score c2[n] - 2*dot (row-constant |x|^2 doesn't move the
    // argmin). Invalid columns carry +BIG. Min-reduce all 8 rows per step so
    // the 8 independent DS chains overlap.
    float sc[8], s[8];
    #pragma unroll
    for (int j = 0; j < 8; ++j) { sc[j] = c2lane - 2.0f * acc[j]; s[j] = sc[j]; }
    {
      float o[8];
      #pragma unroll
      for (int j = 0; j < 8; ++j) o[j] = swz_f<1>(s[j]);
      #pragma unroll
      for (int j = 0; j < 8; ++j) s[j] = fminf(s[j], o[j]);
      #pragma unroll
      for (int j = 0; j < 8; ++j) o[j] = swz_f<2>(s[j]);
      #pragma unroll
      for (int j = 0; j < 8; ++j) s[j] = fminf(s[j], o[j]);
      #pragma unroll
      for (int j = 0; j < 8; ++j) o[j] = swz_f<4>(s[j]);
      #pragma unroll
      for (int j = 0; j < 8; ++j) s[j] = fminf(s[j], o[j]);
      #pragma unroll
      for (int j = 0; j < 8; ++j) o[j] = swz_f<8>(s[j]);
      #pragma unroll
      for (int j = 0; j < 8; ++j) s[j] = fminf(s[j], o[j]);
    }

    #pragma unroll
    for (int j = 0; j < 8; ++j) {
      // Lowest-index argmin via ballot restricted to this lane's half:
      // lowest set bit == lowest center column == jnp.argmin tie-break.
      const unsigned m32  = (unsigned)__ballot(valid && sc[j] == s[j]);
      const unsigned half = hi ? (m32 >> 16) : (m32 & 0xFFFFu);
      const int cand = half ? (__ffs(half) - 1) : MAXC;

      if (n == j) {                        // lanes j and 16+j are the writers
        const int myrow = base + j + 8 * hi;   // C/D row layout
        float dist;
        if ((s_mutbits[myrow >> 5] >> (myrow & 31)) & 1u) {
          dist = 0.0f;                     // mutation nodes forced to 0
        } else if (cand >= MAXC) {
          dist = BIGF;                     // no centers (degenerate)
        } else {
          // exact recompute, matching reference numerics
          const float* q = pos + (size_t)(gbase + myrow) * 3;
          float dx = q[0] - s_cx[cand];
          float dy = q[1] - s_cy[cand];
          float dz = q[2] - s_cz[cand];
          dist = dx * dx + dy * dy + dz * dz;
        }
        s_dist[myrow] = dist;
      }
    }
  }
  __syncthreads();

  // ---- Phase C: exact top-KSEL smallest via 4-pass radix select ----------
  // dist >= 0, so float bits compare as unsigned. After 4 passes: prefix is
  // the K-th smallest bit pattern T, want = how many elements equal to T get
  // selected (lowest indices first => jax.lax.top_k tie-break).
  unsigned prefix = 0;
  int want = KSEL;
  for (int shift = 24; shift >= 0; shift -= 8) {
    s_hist[tid] = 0;                       // BLOCK == 256 covers the table
    __syncthreads();
    #pragma unroll
    for (int e = 0; e < NPG / BLOCK; ++e) {
      const int r = tid * (NPG / BLOCK) + e;
      const unsigned u = __float_as_uint(s_dist[r]);
      const bool cand = (shift == 24) || ((u >> (shift + 8)) == prefix);
      if (cand) atomicAdd(&s_hist[(u >> shift) & 255u], 1u);
    }
    __syncthreads();
    if (tid == 0) {
      unsigned cum = 0; int b = 0;
      for (; b < 256; ++b) {
        const unsigned h = s_hist[b];
        if (cum + h >= (unsigned)want) break;
        cum += h;
      }
      s_bcast[0] = (unsigned)b;
      s_bcast[1] = (unsigned)want - cum;
    }
    __syncthreads();
    prefix = (prefix << 8) | s_bcast[0];
    want   = (int)s_bcast[1];
    __syncthreads();
  }
  const unsigned T = prefix;
  const int neq = want;                    // #(== T) to take, by lowest index

  // rank elements equal to T by index (blocked layout => index order)
  unsigned cnt = 0;
  #pragma unroll
  for (int e = 0; e < NPG / BLOCK; ++e) {
    const int r = tid * (NPG / BLOCK) + e;
    if (__float_as_uint(s_dist[r]) == T) cnt++;
  }
  s_scan[tid] = cnt;
  __syncthreads();
  if (tid == 0) {                          // tiny serial exclusive scan (256)
    unsigned run = 0;
    for (int i = 0; i < BLOCK; ++i) { unsigned c = s_scan[i]; s_scan[i] = run; run += c; }
  }
  __syncthreads();
  unsigned rank = s_scan[tid];
  #pragma unroll
  for (int e = 0; e < NPG / BLOCK; ++e) {
    const int r = tid * (NPG / BLOCK) + e;
    const unsigned u = __float_as_uint(s_dist[r]);
    const bool sel = (u < T) || (u == T && (int)rank < neq);
    if (u == T) rank++;
    out[gbase + r] = sel ? 1.0f : 0.0f;
  }
}

// ---------------------------------------------------------------------------
// Inputs (setup_inputs order): 0 node_position f32[N*3], 1 is_mutation bool[N]
// (assumed raw 1-byte numpy bools), 2 atom_name i32[N], 3 node2graph i32[N]
// (unused: graphs are contiguous, equal-sized), 4 num_nodes i32[G], 5 ca_id.
// Output: node_mask as f32[N] (0.0 / 1.0).
// ---------------------------------------------------------------------------
extern "C" void kernel_launch(void* const* d_in, const int* in_sizes, int n_in,
                              void* d_out, int out_size, void* d_ws, size_t ws_size,
                              hipStream_t stream) {
  const float*   posp  = (const float*)d_in[0];
  const uint8_t* mutp  = (const uint8_t*)d_in[1];
  const int*     atomp = (const int*)d_in[2];
  const int*     ca    = (const int*)d_in[5];
  float*         outp  = (float*)d_out;
  const int G = in_sizes[4];               // num_nodes has one entry per graph
  knn_mutation_kernel<<<G, BLOCK, 0, stream>>>(posp, mutp, atomp, ca, outp);
}